// AutoEncoder_21311627723335
// MI455X (gfx1250) — compile-verified
//
#include <hip/hip_runtime.h>
#include <math.h>

// ---------------------------------------------------------------------------
// Batch-1 2-layer LSTM step + linear head for MI455X (gfx1250, wave32).
// Memory-bound fp32 GEMV (1.6 GB of weights => ~69us @ 23.3 TB/s roofline).
// V_WMMA_F32_16X16X4_F32 is the FMA engine; weights reach the WMMA A-layout
// via coalesced 128b global loads -> padded LDS staging -> ds_load_b64.
// Weights go global->LDS directly with GLOBAL_LOAD_ASYNC_TO_LDS_B128
// (ASYNCcnt, double-buffered); VGPR-staging fallback kept under #if.
// Split-K over the 8 waves of a 256-thread block, deterministic LDS reduce.
// ---------------------------------------------------------------------------

typedef float v2f __attribute__((ext_vector_type(2)));
typedef float v8f __attribute__((ext_vector_type(8)));
typedef int   v4i __attribute__((vector_size(16)));   // matches builtin param

#define AS1 __attribute__((address_space(1)))
#define AS3 __attribute__((address_space(3)))

#define NWAVES 8
#define TPB (NWAVES * 32)
#define CH 32              // K-chunk per pipeline stage (floats)
#define STRIDE 36          // padded LDS row stride (words): bank-conflict-free,
                           // keeps all 128b LDS accesses 16B aligned
#define WBUF (16 * STRIDE) // staging floats per buffer
#define WSTG (2 * WBUF)    // two buffers per wave (async double-buffering)

#if defined(__gfx1250__) && __has_builtin(__builtin_amdgcn_global_load_async_to_lds_b128)
#define USE_ASYNC_LDS 1
#else
#define USE_ASYNC_LDS 0
#endif

#if defined(__gfx1250__) && __has_builtin(__builtin_amdgcn_s_wait_asynccnt)
#define WAIT_ASYNC(n) __builtin_amdgcn_s_wait_asynccnt(n)
#else
#define WAIT_ASYNC(n) asm volatile("s_wait_asynccnt %0" ::"i"(n) : "memory")
#endif

__device__ __forceinline__ v8f wmma_f32_4(v2f a, v2f b, v8f c) {
    return __builtin_amdgcn_wmma_f32_16x16x4_f32(false, a, false, b, (short)0, c,
                                                 false, false);
}

// y[row_base..row_base+15] = (acc ? y : 0) + W[16 rows, K] * v + bias0 + bias1
__global__ __launch_bounds__(TPB) void gemv16_wmma(
    float* __restrict__ y,
    const float* __restrict__ W,
    const float* __restrict__ v,
    int K,
    const float* __restrict__ bias0,
    const float* __restrict__ bias1,
    int accumulate)
{
    extern __shared__ float smem[];
    float* vs   = smem;                       // K floats: staged input vector
    float* wst  = smem + K;                   // NWAVES * WSTG: weight staging
    float* part = wst + NWAVES * WSTG;        // NWAVES*16: split-K partials

    const int tid  = threadIdx.x;
    const int w    = tid >> 5;            // wave id 0..7
    const int lane = tid & 31;
    const int hf   = lane >> 4;           // lane half: K pair {2hf,2hf+1} of each 4-K step for A/B
    const int m    = lane & 15;           // row within the 16-row tile
    const int row_base = blockIdx.x * 16;

    // Stage the input vector once (coalesced); read 8x per block afterwards.
    for (int i = tid; i < K; i += TPB) vs[i] = v[i];
    __syncthreads();

    // Per-wave K segment, a multiple of CH (K is a multiple of 4).
    const int seg = ((K + NWAVES * CH - 1) / (NWAVES * CH)) * CH;
    const int k0  = __builtin_amdgcn_readfirstlane(w * seg);
    const int k1  = __builtin_amdgcn_readfirstlane(min(k0 + seg, K));
    const int nch = __builtin_amdgcn_readfirstlane((k1 - k0) >> 5); // full chunks
    const int rem = __builtin_amdgcn_readfirstlane(k1 - k0 - (nch << 5));

    // Coalesced loader geometry: lane covers (row r4 + 4i, cols c4..c4+3).
    const int r4 = lane >> 3;             // 0..3
    const int c4 = (lane & 7) << 2;       // 0,4,...,28
    const float* gp = W + (size_t)(row_base + r4) * (size_t)K + c4;
    const size_t rK4 = 4 * (size_t)K;     // 4-row stride in floats

    float* buf0 = wst + w * WSTG;         // wave-private staging (no barriers)
    const int so = r4 * STRIDE + c4;      // LDS store offset (floats)
    const int ao = m * STRIDE + 2 * hf;   // LDS A-load offset (floats)

    v8f acc = {0.f, 0.f, 0.f, 0.f, 0.f, 0.f, 0.f, 0.f};

#if USE_ASYNC_LDS
    // -------- async global->LDS path (ASYNCcnt), double buffered --------
    float* buf1 = buf0 + WBUF;
    auto issue4 = [&](const float* g, float* b) {
        __builtin_amdgcn_global_load_async_to_lds_b128(
            (AS1 v4i*)(g),            (AS3 v4i*)(b + so), 0, 0);
        __builtin_amdgcn_global_load_async_to_lds_b128(
            (AS1 v4i*)(g + rK4),      (AS3 v4i*)(b + so + 4 * STRIDE), 0, 0);
        __builtin_amdgcn_global_load_async_to_lds_b128(
            (AS1 v4i*)(g + 2 * rK4),  (AS3 v4i*)(b + so + 8 * STRIDE), 0, 0);
        __builtin_amdgcn_global_load_async_to_lds_b128(
            (AS1 v4i*)(g + 3 * rK4),  (AS3 v4i*)(b + so + 12 * STRIDE), 0, 0);
    };

    if (nch > 0) {
        issue4(gp + k0, buf0);
        int kc = k0;
        for (int c = 0; c < nch; ++c) {
            float* cur = (c & 1) ? buf1 : buf0;
            float* nxt = (c & 1) ? buf0 : buf1;
            if (c + 1 < nch) {
                // All ds_loads of the buffer we are about to overwrite have
                // completed (their consumers already waited), but make the
                // ordering explicit before async writes reuse it.
                asm volatile("s_wait_dscnt 0x0" ::: "memory");
                issue4(gp + kc + CH, nxt);
                WAIT_ASYNC(4);   // current chunk's 4 loads retired (in-order)
            } else {
                WAIT_ASYNC(0);   // last chunk: everything written to LDS
            }
            asm volatile("" ::: "memory");
            #pragma unroll
            for (int t = 0; t < 8; ++t) {
                v2f a = *(const v2f*)(cur + ao + 4 * t);
                v2f b = *(const v2f*)(vs + kc + 4 * t + 2 * hf); // broadcast
                acc = wmma_f32_4(a, b, acc);
            }
            kc += CH;
        }
    }
#else
    // -------- fallback: global->VGPR->LDS staging, software pipelined --------
    float* buf = buf0;
    if (nch > 0) {
        const float* g = gp + k0;
        float4 p0 = *(const float4*)(g);
        float4 p1 = *(const float4*)(g + rK4);
        float4 p2 = *(const float4*)(g + 2 * rK4);
        float4 p3 = *(const float4*)(g + 3 * rK4);

        int kc = k0;
        for (int c = 0; c < nch; ++c) {
            *(float4*)(buf + so)               = p0;
            *(float4*)(buf + so + 4 * STRIDE)  = p1;
            *(float4*)(buf + so + 8 * STRIDE)  = p2;
            *(float4*)(buf + so + 12 * STRIDE) = p3;

            if (c + 1 < nch) {
                const float* gn = gp + kc + CH;
                p0 = *(const float4*)(gn);
                p1 = *(const float4*)(gn + rK4);
                p2 = *(const float4*)(gn + 2 * rK4);
                p3 = *(const float4*)(gn + 3 * rK4);
            }

            #pragma unroll
            for (int t = 0; t < 8; ++t) {
                v2f a = *(const v2f*)(buf + ao + 4 * t);
                v2f b = *(const v2f*)(vs + kc + 4 * t + 2 * hf); // broadcast
                acc = wmma_f32_4(a, b, acc);
            }
            kc += CH;
        }
    }
#endif

    // Remainder tiles (<CH, multiple of 4; only the last wave, K=5000 case):
    // tiny traffic, simple strided path.
    if (rem > 0) {
        const float* aptr = W + (size_t)(row_base + m) * (size_t)K + 2 * hf;
        const int kk = k0 + (nch << 5);
        for (int k = kk; k < kk + rem; k += 4) {
            v2f a = *(const v2f*)(aptr + k);
            v2f b = *(const v2f*)(vs + k + 2 * hf);
            acc = wmma_f32_4(a, b, acc);
        }
    }

    // Every D column equals the result; column 0 lives in lanes 0 (M=0..7 in
    // acc[0..7]) and 16 (M=8..15).
    if ((lane & 15) == 0) {
        float* dst = &part[w * 16 + 8 * hf];
        *(float4*)(dst)     = make_float4(acc[0], acc[1], acc[2], acc[3]);
        *(float4*)(dst + 4) = make_float4(acc[4], acc[5], acc[6], acc[7]);
    }
    __syncthreads();

    // Deterministic fixed-order reduction over the 8 wave partials.
    if (tid < 16) {
        float s = 0.f;
        #pragma unroll
        for (int ww = 0; ww < NWAVES; ++ww) s += part[ww * 16 + tid];
        const int r = row_base + tid;
        if (bias0) s += bias0[r];
        if (bias1) s += bias1[r];
        if (accumulate) s += y[r];
        y[r] = s;
    }
}

// Elementwise LSTM cell: gates layout [i | f | g | o], each H wide.
__global__ void lstm_cell_kernel(const float* __restrict__ gates,
                                 const float* __restrict__ c_prev,
                                 float* __restrict__ h_out,
                                 float* __restrict__ c_out,
                                 int H)
{
    int j = blockIdx.x * blockDim.x + threadIdx.x;
    if (j >= H) return;
    float i = 1.f / (1.f + expf(-gates[j]));
    float f = 1.f / (1.f + expf(-gates[H + j]));
    float g = tanhf(gates[2 * H + j]);
    float o = 1.f / (1.f + expf(-gates[3 * H + j]));
    float c = f * c_prev[j] + i * g;
    c_out[j] = c;
    h_out[j] = o * tanhf(c);
}

extern "C" void kernel_launch(void* const* d_in, const int* in_sizes, int n_in,
                              void* d_out, int out_size, void* d_ws, size_t ws_size,
                              hipStream_t stream) {
    const float* x     = (const float*)d_in[0];
    const float* w_ih1 = (const float*)d_in[1];
    const float* w_hh1 = (const float*)d_in[2];
    const float* b_ih1 = (const float*)d_in[3];
    const float* b_hh1 = (const float*)d_in[4];
    const float* w_ih2 = (const float*)d_in[5];
    const float* w_hh2 = (const float*)d_in[6];
    const float* b_ih2 = (const float*)d_in[7];
    const float* b_hh2 = (const float*)d_in[8];
    const float* w_lin = (const float*)d_in[9];
    const float* b_lin = (const float*)d_in[10];
    const float* h_t   = (const float*)d_in[11];
    const float* c_t   = (const float*)d_in[12];
    const float* h2_t  = (const float*)d_in[13];
    const float* c2_t  = (const float*)d_in[14];

    const int I = 4000, H = 5000, G = 4 * H;   // G = 20000 gate rows

    float* ws     = (float*)d_ws;
    float* gates1 = ws;               // G
    float* gates2 = ws + G;           // G
    float* h1     = ws + 2 * G;       // H
    float* c1     = h1 + H;           // H
    float* h2     = c1 + H;           // H
    float* c2     = h2 + H;           // H

    // LDS layout sized for the (larger) double-buffered async path, used by
    // both device code paths (host pass cannot see which one compiled in).
    const size_t stage = (size_t)(NWAVES * WSTG + NWAVES * 16) * 4;
    const size_t smemI = (size_t)I * 4 + stage;
    const size_t smemH = (size_t)H * 4 + stage;

    // gates1 = x*W_ih1^T + b_ih1 + h_t*W_hh1^T + b_hh1
    gemv16_wmma<<<G / 16, TPB, smemI, stream>>>(gates1, w_ih1, x,   I, b_ih1, b_hh1, 0);
    gemv16_wmma<<<G / 16, TPB, smemH, stream>>>(gates1, w_hh1, h_t, H, nullptr, nullptr, 1);
    // gates2 partial = h2_t*W_hh2^T + b_ih2 + b_hh2   (independent of layer 1)
    gemv16_wmma<<<G / 16, TPB, smemH, stream>>>(gates2, w_hh2, h2_t, H, b_ih2, b_hh2, 0);
    // layer-1 cell -> h1
    lstm_cell_kernel<<<(H + 255) / 256, 256, 0, stream>>>(gates1, c_t, h1, c1, H);
    // gates2 += h1*W_ih2^T
    gemv16_wmma<<<G / 16, TPB, smemH, stream>>>(gates2, w_ih2, h1, H, nullptr, nullptr, 1);
    // layer-2 cell -> h2
    lstm_cell_kernel<<<(H + 255) / 256, 256, 0, stream>>>(gates2, c2_t, h2, c2, H);
    // output = h2*W_lin^T + b_lin
    gemv16_wmma<<<I / 16, TPB, smemH, stream>>>((float*)d_out, w_lin, h2, H, b_lin, nullptr, 0);
}